// MultiBoxLoss_29411936043747
// MI455X (gfx1250) — compile-verified
//
#include <hip/hip_runtime.h>
#include <stdint.h>

#define Bb 32
#define Aa 16384
#define Mm 64
#define Nn (Bb*Aa)          // 524288
#define NBLK (Nn/256)       // 2048
#define HISTBLK 512

// ---------------- workspace layout (byte offsets) ----------------
#define OFF_OVERLAP  ((size_t)0)                       // Nn floats
#define OFF_DIGIT    (OFF_OVERLAP + (size_t)Nn*4)      // Nn ints
#define OFF_NEGV     (OFF_DIGIT   + (size_t)Nn*4)      // Nn floats
#define OFF_BESTA    (OFF_NEGV    + (size_t)Nn*4)      // Bb*Mm ints
#define OFF_HIST     (OFF_BESTA   + (size_t)Bb*Mm*4)   // 4*256 u32
#define OFF_SMPART   (OFF_HIST    + (size_t)4*256*4)   // NBLK floats
#define OFF_POSPART  (OFF_SMPART  + (size_t)NBLK*4)    // NBLK floats
#define OFF_GTPART   (OFF_POSPART + (size_t)NBLK*4)    // NBLK floats
#define OFF_SCAL     (OFF_GTPART  + (size_t)NBLK*4)    // [0]=total [1]=count_gt [2]=prefix [3]=kk

typedef __attribute__((ext_vector_type(2))) float v2f;
typedef __attribute__((ext_vector_type(8))) float v8f;
typedef int v4i __attribute__((vector_size(16)));      // matches builtin param type
typedef __attribute__((address_space(1))) v4i* gv4p;   // global int4*
typedef __attribute__((address_space(3))) v4i* lv4p;   // LDS int4*

#if defined(__has_builtin)
#if __has_builtin(__builtin_amdgcn_wmma_f32_16x16x4_f32)
#define HAVE_WMMA 1
#endif
#if __has_builtin(__builtin_amdgcn_global_load_async_to_lds_b128)
#define HAVE_ASYNC 1
#endif
#endif
#ifndef HAVE_WMMA
#define HAVE_WMMA 0
#endif
#ifndef HAVE_ASYNC
#define HAVE_ASYNC 0
#endif

// ---- exact 32-lane f32 sum via V_WMMA_F32_16X16X4_F32 with B = ones.
// A[m,k]: lane<16 -> A[lane,0]=v, lane>=16 -> A[lane-16,2]=v, other K slots 0.
// D[m,n] = v_m + v_{m+16}; lane in first half sums D vgprs j=0..7 -> sum(v0..7,v16..23),
// second half -> sum(v8..15,v24..31); one xor-16 shuffle adds halves. Exact (all f32 adds).
__device__ __forceinline__ float wave_sum32(float v) {
#if HAVE_WMMA
  v2f a, bm; v8f c;
  float* ap = (float*)&a; float* bp = (float*)&bm; float* cp = (float*)&c;
  ap[0] = v;    ap[1] = 0.0f;
  bp[0] = 1.0f; bp[1] = 1.0f;
#pragma unroll
  for (int j = 0; j < 8; ++j) cp[j] = 0.0f;
  c = __builtin_amdgcn_wmma_f32_16x16x4_f32(false, a, false, bm, (short)0, c, false, false);
  float s = 0.0f;
#pragma unroll
  for (int j = 0; j < 8; ++j) s += cp[j];
  s += __shfl_xor(s, 16, 32);
  return s;
#else
  for (int off = 16; off > 0; off >>= 1) v += __shfl_xor(v, off, 32);
  return v;
#endif
}

// stage 64 boxes (256 floats = 1KB) into LDS; async-to-LDS on CDNA5
__device__ __forceinline__ void stage_boxes(const float* __restrict__ src, float* sbox) {
#if HAVE_ASYNC
  if (threadIdx.x < 32) {
    int t = threadIdx.x;
    {
      gv4p gp = (gv4p)(uintptr_t)(src + t * 4);
      lv4p lp = (lv4p)(uint32_t)(uintptr_t)(sbox + t * 4);
      __builtin_amdgcn_global_load_async_to_lds_b128(gp, lp, 0, 0);
    }
    {
      gv4p gp = (gv4p)(uintptr_t)(src + 128 + t * 4);
      lv4p lp = (lv4p)(uint32_t)(uintptr_t)(sbox + 128 + t * 4);
      __builtin_amdgcn_global_load_async_to_lds_b128(gp, lp, 0, 0);
    }
    asm volatile("s_wait_asynccnt 0x0" ::: "memory");
  }
  __syncthreads();
#else
  sbox[threadIdx.x] = src[threadIdx.x];   // blockDim == 256 == 64*4
  __syncthreads();
#endif
}

__device__ __forceinline__ float iou_pair(float bx1, float by1, float bx2, float by2, float areaA,
                                          float dx1, float dy1, float dx2, float dy2, float areaB) {
  float ltx = fmaxf(bx1, dx1), lty = fmaxf(by1, dy1);
  float rbx = fminf(bx2, dx2), rby = fminf(by2, dy2);
  float w = fmaxf(rbx - ltx, 0.0f), h = fmaxf(rby - lty, 0.0f);
  float inter = w * h;
  return inter / (areaA + areaB - inter);
}

// ---------------- kernels ----------------
__global__ void k_init(uint32_t* hist, int* scal) {
  int t = blockIdx.x * 256 + threadIdx.x;
  if (t < 1024) hist[t] = 0u;
  if (t < 4) scal[t] = 0;
}

// per-anchor max/argmax over M boxes
__global__ __launch_bounds__(256) void k_match(const float* __restrict__ boxes,
                                               const float* __restrict__ defs,
                                               float* __restrict__ overlap,
                                               int* __restrict__ digit) {
  __shared__ float sbox[Mm * 4];
  int b = blockIdx.x >> 6;
  int a = ((blockIdx.x & 63) << 8) + threadIdx.x;
  stage_boxes(boxes + (size_t)b * Mm * 4, sbox);

  float dcx = defs[a * 4 + 0], dcy = defs[a * 4 + 1];
  float dw  = defs[a * 4 + 2], dh  = defs[a * 4 + 3];
  float dx1 = dcx - dw / 2.0f, dy1 = dcy - dh / 2.0f;
  float dx2 = dcx + dw / 2.0f, dy2 = dcy + dh / 2.0f;
  float areaB = (dx2 - dx1) * (dy2 - dy1);

  float best = -1.0f; int bm = 0;
  for (int m = 0; m < Mm; ++m) {
    float bx1 = sbox[m * 4 + 0], by1 = sbox[m * 4 + 1];
    float bx2 = sbox[m * 4 + 2], by2 = sbox[m * 4 + 3];
    float areaA = (bx2 - bx1) * (by2 - by1);
    float iou = iou_pair(bx1, by1, bx2, by2, areaA, dx1, dy1, dx2, dy2, areaB);
    if (iou > best) { best = iou; bm = m; }   // strict > keeps first index (jnp.argmax)
  }
  overlap[b * Aa + a] = best;
  digit[b * Aa + a] = bm;
}

// per-object argmax over all anchors (one block per (b,m); tie-break = smallest a)
__global__ __launch_bounds__(256) void k_bestanchor(const float* __restrict__ boxes,
                                                    const float* __restrict__ defs,
                                                    int* __restrict__ besta) {
  __shared__ uint64_t sk[256];
  int bmid = blockIdx.x;           // b*Mm + m
  int b = bmid >> 6, m = bmid & 63;
  float bx1 = boxes[(size_t)(b * Mm + m) * 4 + 0];
  float by1 = boxes[(size_t)(b * Mm + m) * 4 + 1];
  float bx2 = boxes[(size_t)(b * Mm + m) * 4 + 2];
  float by2 = boxes[(size_t)(b * Mm + m) * 4 + 3];
  float areaA = (bx2 - bx1) * (by2 - by1);

  uint64_t key = 0;
  for (int a = threadIdx.x; a < Aa; a += 256) {
    float dcx = defs[a * 4 + 0], dcy = defs[a * 4 + 1];
    float dw  = defs[a * 4 + 2], dh  = defs[a * 4 + 3];
    float dx1 = dcx - dw / 2.0f, dy1 = dcy - dh / 2.0f;
    float dx2 = dcx + dw / 2.0f, dy2 = dcy + dh / 2.0f;
    float areaB = (dx2 - dx1) * (dy2 - dy1);
    float iou = iou_pair(bx1, by1, bx2, by2, areaA, dx1, dy1, dx2, dy2, areaB);
    uint64_t k = (((uint64_t)__float_as_uint(iou)) << 32) | (uint32_t)(0xFFFFFFFFu - (uint32_t)a);
    if (k > key) key = k;
  }
  sk[threadIdx.x] = key;
  __syncthreads();
  for (int s = 128; s > 0; s >>= 1) {
    if (threadIdx.x < s && sk[threadIdx.x + s] > sk[threadIdx.x]) sk[threadIdx.x] = sk[threadIdx.x + s];
    __syncthreads();
  }
  if (threadIdx.x == 0)
    besta[bmid] = (int)(0xFFFFFFFFu - (uint32_t)(sk[0] & 0xFFFFFFFFull));
}

// sequential forced-match scatter (last write wins, matches XLA scatter)
__global__ void k_fixup(const int* __restrict__ besta, int* __restrict__ digit,
                        float* __restrict__ overlap) {
  int b = threadIdx.x;
  if (b < Bb) {
    for (int m = 0; m < Mm; ++m) {
      int ba = besta[b * Mm + m];
      digit[b * Aa + ba] = m;
      overlap[b * Aa + ba] = 1.0f;
    }
  }
}

// per-anchor loss terms: count positives, smooth-L1 partials, pos-entropy partials, negv array
__global__ __launch_bounds__(256) void k_loss(const float* __restrict__ x, const float* __restrict__ y,
                                              const float* __restrict__ boxes, const int* __restrict__ labels,
                                              const float* __restrict__ defs, const float* __restrict__ overlap,
                                              const int* __restrict__ digit, float* __restrict__ negv,
                                              float* __restrict__ sm_part, float* __restrict__ pos_part,
                                              int* __restrict__ total) {
  __shared__ float sbox[Mm * 4];
  __shared__ float red[8];
  int b = blockIdx.x >> 6;
  int a = ((blockIdx.x & 63) << 8) + threadIdx.x;
  int i = b * Aa + a;
  stage_boxes(boxes + (size_t)b * Mm * 4, sbox);

  float ov = overlap[i];
  int dg = digit[i];
  int glab = labels[b * Mm + dg];
  int cls = (ov < 0.5f) ? 0 : glab;
  bool pos = (cls == 1);

  // localization target: xyxy -> cxcywh -> gcxgcy
  float bx1 = sbox[dg * 4 + 0], by1 = sbox[dg * 4 + 1];
  float bx2 = sbox[dg * 4 + 2], by2 = sbox[dg * 4 + 3];
  float cx = (bx1 + bx2) / 2.0f, cy = (by1 + by2) / 2.0f;
  float w = bx2 - bx1, h = by2 - by1;
  float dcx = defs[a * 4 + 0], dcy = defs[a * 4 + 1];
  float dw  = defs[a * 4 + 2], dh  = defs[a * 4 + 3];
  float g0 = (cx - dcx) / (dw / 10.0f);
  float g1 = (cy - dcy) / (dh / 10.0f);
  float g2 = logf(w / dw) * 5.0f;
  float g3 = logf(h / dh) * 5.0f;

  float sm = 0.0f;
  if (pos) {
    float d0 = x[(size_t)i * 4 + 0] - g0;
    float d1 = x[(size_t)i * 4 + 1] - g1;
    float d2 = x[(size_t)i * 4 + 2] - g2;
    float d3 = x[(size_t)i * 4 + 3] - g3;
    float a0 = fabsf(d0), a1 = fabsf(d1), a2 = fabsf(d2), a3 = fabsf(d3);
    sm += (a0 < 1.0f) ? 0.5f * d0 * d0 : a0 - 0.5f;
    sm += (a1 < 1.0f) ? 0.5f * d1 * d1 : a1 - 0.5f;
    sm += (a2 < 1.0f) ? 0.5f * d2 * d2 : a2 - 0.5f;
    sm += (a3 < 1.0f) ? 0.5f * d3 * d3 : a3 - 0.5f;
  }

  // 2-class log-softmax cross-entropy
  float y0 = y[(size_t)i * 2 + 0], y1 = y[(size_t)i * 2 + 1];
  float mx = fmaxf(y0, y1);
  float lse = mx + logf(expf(y0 - mx) + expf(y1 - mx));
  float ent = lse - ((cls != 0) ? y1 : y0);
  float posv = pos ? ent : 0.0f;
  negv[i] = pos ? 0.0f : ent;

  unsigned long long blt = __ballot(pos);
  if ((threadIdx.x & 31) == 0) atomicAdd(total, (int)__popcll(blt));

  int wv = threadIdx.x >> 5, ln = threadIdx.x & 31;
  float s1 = wave_sum32(sm);
  if (ln == 0) red[wv] = s1;
  __syncthreads();
  if (threadIdx.x == 0) { float t = 0.0f; for (int q = 0; q < 8; ++q) t += red[q]; sm_part[blockIdx.x] = t; }
  __syncthreads();
  float s2 = wave_sum32(posv);
  if (ln == 0) red[wv] = s2;
  __syncthreads();
  if (threadIdx.x == 0) { float t = 0.0f; for (int q = 0; q < 8; ++q) t += red[q]; pos_part[blockIdx.x] = t; }
}

// 8-bit radix histogram pass over nonneg-float bit patterns
__global__ __launch_bounds__(256) void k_hist(const float* __restrict__ negv, uint32_t* __restrict__ hist,
                                              const uint32_t* __restrict__ scal, int pass) {
  __shared__ uint32_t lh[256];
  lh[threadIdx.x] = 0u;
  __syncthreads();
  uint32_t prefix = (pass == 0) ? 0u : scal[2];
  int shift = 24 - pass * 8;
  uint32_t mask_hi = (pass == 0) ? 0u : (0xFFFFFFFFu << (shift + 8));
  for (int i = blockIdx.x * 256 + threadIdx.x; i < Nn; i += gridDim.x * 256) {
    uint32_t u = __float_as_uint(negv[i]);
    if ((u & mask_hi) == prefix) atomicAdd(&lh[(u >> shift) & 0xFFu], 1u);
  }
  __syncthreads();
  uint32_t v = lh[threadIdx.x];
  if (v) atomicAdd(&hist[pass * 256 + threadIdx.x], v);
}

// pick the bin containing the kk-th largest; refine prefix
__global__ void k_select(const uint32_t* __restrict__ hist, uint32_t* __restrict__ scal,
                         const int* __restrict__ total, int pass) {
  if (blockIdx.x == 0 && threadIdx.x == 0) {
    uint32_t kk = (pass == 0) ? (uint32_t)(*total) : scal[3];
    uint32_t prefix = (pass == 0) ? 0u : scal[2];
    int shift = 24 - pass * 8;
    uint32_t cum = 0;
    for (int bin = 255; bin >= 0; --bin) {
      uint32_t hc = hist[pass * 256 + bin];
      if (cum + hc >= kk) { prefix |= ((uint32_t)bin) << shift; kk -= cum; break; }
      cum += hc;
    }
    scal[2] = prefix;   // bits of the k-th largest value
    scal[3] = kk;
  }
}

// sum & count of values strictly greater than threshold
__global__ __launch_bounds__(256) void k_sumgt(const float* __restrict__ negv,
                                               const uint32_t* __restrict__ scal,
                                               float* __restrict__ gt_part, int* __restrict__ count_gt) {
  __shared__ float red[8];
  uint32_t thr = scal[2];
  int i = blockIdx.x * 256 + threadIdx.x;
  float v = negv[i];
  uint32_t u = __float_as_uint(v);
  bool gt = (u > thr);
  float s = gt ? v : 0.0f;
  unsigned long long blt = __ballot(gt);
  if ((threadIdx.x & 31) == 0) atomicAdd(count_gt, (int)__popcll(blt));
  float ws = wave_sum32(s);
  int wv = threadIdx.x >> 5, ln = threadIdx.x & 31;
  if (ln == 0) red[wv] = ws;
  __syncthreads();
  if (threadIdx.x == 0) { float t = 0.0f; for (int q = 0; q < 8; ++q) t += red[q]; gt_part[blockIdx.x] = t; }
}

// deterministic final combine
__global__ __launch_bounds__(256) void k_final(const float* __restrict__ sm_part,
                                               const float* __restrict__ pos_part,
                                               const float* __restrict__ gt_part,
                                               const uint32_t* __restrict__ scal,
                                               const int* __restrict__ total,
                                               const int* __restrict__ count_gt,
                                               float* __restrict__ out) {
  __shared__ float sh[256];
  const float* arr[3] = { sm_part, pos_part, gt_part };
  float res[3];
  for (int q = 0; q < 3; ++q) {
    float acc = 0.0f;
    for (int j = threadIdx.x; j < NBLK; j += 256) acc += arr[q][j];
    sh[threadIdx.x] = acc;
    __syncthreads();
    for (int s = 128; s > 0; s >>= 1) {
      if (threadIdx.x < s) sh[threadIdx.x] += sh[threadIdx.x + s];
      __syncthreads();
    }
    res[q] = sh[0];
    __syncthreads();
  }
  if (threadIdx.x == 0) {
    int tot = *total;
    float totf = (float)tot;
    int keep = (int)(totf * 1.0f);             // RATIO=1 -> == tot (tot < 2^24, exact)
    float vth = __uint_as_float(scal[2]);      // exact k-th largest value
    float neg_sum = res[2] + (float)(keep - *count_gt) * vth;
    float smooth = res[0] / (totf * 4.0f);
    float entropy = (res[1] + neg_sum) / totf; // ALPHA=1
    out[0] = smooth + entropy;
  }
}

extern "C" void kernel_launch(void* const* d_in, const int* in_sizes, int n_in,
                              void* d_out, int out_size, void* d_ws, size_t ws_size,
                              hipStream_t stream) {
  const float* x      = (const float*)d_in[0];
  const float* y      = (const float*)d_in[1];
  const float* boxes  = (const float*)d_in[2];
  const int*   labels = (const int*)d_in[3];
  const float* defs   = (const float*)d_in[4];

  uint8_t* ws = (uint8_t*)d_ws;
  float*    overlap = (float*)(ws + OFF_OVERLAP);
  int*      digit   = (int*)(ws + OFF_DIGIT);
  float*    negv    = (float*)(ws + OFF_NEGV);
  int*      besta   = (int*)(ws + OFF_BESTA);
  uint32_t* hist    = (uint32_t*)(ws + OFF_HIST);
  float*    smp     = (float*)(ws + OFF_SMPART);
  float*    psp     = (float*)(ws + OFF_POSPART);
  float*    gtp     = (float*)(ws + OFF_GTPART);
  uint32_t* scal    = (uint32_t*)(ws + OFF_SCAL);
  int*      total   = (int*)&scal[0];
  int*      cgt     = (int*)&scal[1];

  k_init<<<4, 256, 0, stream>>>(hist, (int*)scal);
  k_match<<<NBLK, 256, 0, stream>>>(boxes, defs, overlap, digit);
  k_bestanchor<<<Bb * Mm, 256, 0, stream>>>(boxes, defs, besta);
  k_fixup<<<1, 32, 0, stream>>>(besta, digit, overlap);
  k_loss<<<NBLK, 256, 0, stream>>>(x, y, boxes, labels, defs, overlap, digit, negv, smp, psp, total);
  for (int p = 0; p < 4; ++p) {
    k_hist<<<HISTBLK, 256, 0, stream>>>(negv, hist, scal, p);
    k_select<<<1, 32, 0, stream>>>(hist, scal, total, p);
  }
  k_sumgt<<<NBLK, 256, 0, stream>>>(negv, scal, gtp, cgt);
  k_final<<<1, 256, 0, stream>>>(smp, psp, gtp, scal, total, cgt, (float*)d_out);
}